// ISS_IFHMM_27805618274977
// MI455X (gfx1250) — compile-verified
//
#include <hip/hip_runtime.h>
#include <hip/hip_bf16.h>
#include <math.h>

// Problem constants
#define TT 2048
#define CC 8
#define SS 16
#define NN 10000

typedef __attribute__((ext_vector_type(2))) float v2f;
typedef __attribute__((ext_vector_type(8))) float v8f;

// ---------------------------------------------------------------------------
// K1: softmax over C=8 chain weights per time step -> probability domain
// ---------------------------------------------------------------------------
__global__ __launch_bounds__(256) void cw_softmax_kernel(const float* __restrict__ x,
                                                         float* __restrict__ y) {
    const int t = blockIdx.x * 256 + threadIdx.x;
    if (t >= TT) return;
    float v[CC];
    float m = -3.4e38f;
#pragma unroll
    for (int c = 0; c < CC; ++c) { v[c] = x[t * CC + c]; m = fmaxf(m, v[c]); }
    float s = 0.f;
#pragma unroll
    for (int c = 0; c < CC; ++c) { v[c] = __expf(v[c] - m); s += v[c]; }
    const float inv = 1.0f / s;
#pragma unroll
    for (int c = 0; c < CC; ++c) y[t * CC + c] = v[c] * inv;
}

// ---------------------------------------------------------------------------
// K2: softmax over N=10000 for each (c,s) emission row -> probability domain
// one block per row, block reduction for max and sum
// ---------------------------------------------------------------------------
__global__ __launch_bounds__(256) void emis_softmax_kernel(const float* __restrict__ x,
                                                           float* __restrict__ y) {
    __shared__ float red[256];
    const int row = blockIdx.x;           // 0 .. C*S-1
    const float* xr = x + (size_t)row * NN;
    float* yr = y + (size_t)row * NN;
    const int tid = threadIdx.x;

    float m = -3.4e38f;
    for (int i = tid; i < NN; i += 256) m = fmaxf(m, xr[i]);
    red[tid] = m;
    __syncthreads();
    for (int s = 128; s > 0; s >>= 1) {
        if (tid < s) red[tid] = fmaxf(red[tid], red[tid + s]);
        __syncthreads();
    }
    m = red[0];
    __syncthreads();

    float sum = 0.f;
    for (int i = tid; i < NN; i += 256) sum += __expf(xr[i] - m);
    red[tid] = sum;
    __syncthreads();
    for (int s = 128; s > 0; s >>= 1) {
        if (tid < s) red[tid] += red[tid + s];
        __syncthreads();
    }
    const float inv = 1.0f / red[0];

    for (int i = tid; i < NN; i += 256) yr[i] = __expf(xr[i] - m) * inv;
}

// ---------------------------------------------------------------------------
// K3: sequential HMM scan in probability domain.
// One block, 128 threads = 4 waves; each wave handles 2 chains
// (lanes 0-15 -> chain 2w, lanes 16-31 -> chain 2w+1), cross-lane via __shfl
// within 16-lane groups -> NO barriers inside the 2048-step loop.
// Writes probability H to workspace and log_hidden output.
// ---------------------------------------------------------------------------
__global__ __launch_bounds__(128) void scan_kernel(const float* __restrict__ init_un,
                                                   const float* __restrict__ A_un,
                                                   float* __restrict__ Hprob,
                                                   float* __restrict__ log_hidden) {
    __shared__ float sA[CC * SS * SS];    // normalized transition probs
    const int tid = threadIdx.x;

    // --- normalize transition matrix: one (c, s_row) softmax per thread ---
    {
        const int cr = tid >> 4;          // chain
        const int sr = tid & 15;          // row (source state)
        const float* ar = A_un + (cr * SS + sr) * SS;
        float v[SS];
        float m = -3.4e38f;
#pragma unroll
        for (int j = 0; j < SS; ++j) { v[j] = ar[j]; m = fmaxf(m, v[j]); }
        float sum = 0.f;
#pragma unroll
        for (int j = 0; j < SS; ++j) { v[j] = __expf(v[j] - m); sum += v[j]; }
        const float inv = 1.0f / sum;
#pragma unroll
        for (int j = 0; j < SS; ++j) sA[(cr * SS + sr) * SS + j] = v[j] * inv;
    }
    __syncthreads();

    const int wave = tid >> 5;
    const int lane = tid & 31;
    const int c = 2 * wave + (lane >> 4); // chain of this lane
    const int s = lane & 15;              // destination state of this lane

    // column A[c, :, s] kept in registers
    float Acol[SS];
#pragma unroll
    for (int j = 0; j < SS; ++j) Acol[j] = sA[(c * SS + j) * SS + s];

    // --- h0 = softmax(init[c, :]) via 16-lane butterfly reduction ---
    const float x = init_un[c * SS + s];
    float m = x;
#pragma unroll
    for (int off = 8; off > 0; off >>= 1) m = fmaxf(m, __shfl_xor(m, off, 16));
    const float e = __expf(x - m);
    float sum = e;
#pragma unroll
    for (int off = 8; off > 0; off >>= 1) sum += __shfl_xor(sum, off, 16);
    float p = e / sum;

    // --- sequential scan: p_t written, then p <- p @ A ---
    for (int t = 0; t < TT; ++t) {
        const int idx = t * (CC * SS) + c * SS + s;
        Hprob[idx] = p;
        log_hidden[idx] = __logf(p);
        float pn = 0.f;
#pragma unroll
        for (int j = 0; j < SS; ++j) pn = fmaf(__shfl(p, j, 16), Acol[j], pn);
        p = pn;
    }
}

// ---------------------------------------------------------------------------
// K4: emit GEMM + logs. For each (t-tile of 16, n-tile of 16, chain c):
//   P[16t x 16n] = H_tile(16x16) @ E_tile(16x16)  via 4x V_WMMA_F32_16X16X4_F32
//   log_obs_ = log(P);  log_obs = log(sum_c cw * P)
// Block: 8 waves (dim 32x8); all waves share the same t-tile (H, cw in LDS),
// each wave owns a distinct n-tile. N = 625*16 exactly, T = 128*16 exactly.
// ---------------------------------------------------------------------------
#define SH_STRIDE 132   // 132 % 64 == 4 -> conflict-free row access
__global__ __launch_bounds__(256) void emit_kernel(const float* __restrict__ E,
                                                   const float* __restrict__ CW,
                                                   const float* __restrict__ H,
                                                   float* __restrict__ log_obs,
                                                   float* __restrict__ log_obs2) {
    __shared__ float sH[16 * SH_STRIDE];  // H[t0..t0+15][c][s], padded
    __shared__ float sCW[16 * CC];
    const int tid = threadIdx.y * 32 + threadIdx.x;
    const int t0 = blockIdx.y * 16;

#pragma unroll
    for (int j = 0; j < 8; ++j) {
        const int e = tid + 256 * j;                 // 0 .. 2047
        sH[(e >> 7) * SH_STRIDE + (e & 127)] = H[t0 * (CC * SS) + e];
    }
    if (tid < 16 * CC) sCW[tid] = CW[t0 * CC + tid];
    __syncthreads();

    const int n_tile = blockIdx.x * 8 + threadIdx.y;
    if (n_tile >= NN / 16) return;       // uniform per wave; EXEC stays all-ones

    const int lane = threadIdx.x;
    const int half = lane >> 4;          // 0: lanes 0-15, 1: lanes 16-31
    const int l15 = lane & 15;
    const int n = n_tile * 16 + l15;     // column this lane loads/stores

    float wsum[8];
#pragma unroll
    for (int r = 0; r < 8; ++r) wsum[r] = 0.f;

    for (int c = 0; c < CC; ++c) {
        v8f acc = {};
#pragma unroll
        for (int k = 0; k < 4; ++k) {
            // A (16x4 f32): lane L holds M=L; half 0 -> K = 4k+{0,1}, half 1 -> K = 4k+{2,3}
            // B (4x16 f32): VGPR r -> K = 4k + r + 2*half, N = l15
            const int kb = 4 * k + 2 * half;
            v2f a, b;
            a.x = sH[l15 * SH_STRIDE + c * SS + kb];
            a.y = sH[l15 * SH_STRIDE + c * SS + kb + 1];
            b.x = E[(size_t)(c * SS + kb) * NN + n];
            b.y = E[(size_t)(c * SS + kb + 1) * NN + n];
            acc = __builtin_amdgcn_wmma_f32_16x16x4_f32(
                false, a, false, b, (short)0, acc, false, false);
        }
        // C/D layout: VGPR r -> row (r + 8*half), col = l15
#pragma unroll
        for (int r = 0; r < 8; ++r) {
            const int row = r + 8 * half;
            const float val = acc[r];
            log_obs2[((size_t)(t0 + row) * CC + c) * NN + n] = __logf(val);
            wsum[r] = fmaf(sCW[row * CC + c], val, wsum[r]);
        }
    }
#pragma unroll
    for (int r = 0; r < 8; ++r) {
        const int row = r + 8 * half;
        log_obs[(size_t)(t0 + row) * NN + n] = __logf(wsum[r]);
    }
}

// ---------------------------------------------------------------------------
extern "C" void kernel_launch(void* const* d_in, const int* in_sizes, int n_in,
                              void* d_out, int out_size, void* d_ws, size_t ws_size,
                              hipStream_t stream) {
    const float* cw_un   = (const float*)d_in[0];  // (T, C)
    const float* init_un = (const float*)d_in[1];  // (C, S)
    const float* E_un    = (const float*)d_in[2];  // (C, S, N)
    const float* A_un    = (const float*)d_in[3];  // (C, S, S)

    float* out = (float*)d_out;
    float* log_obs    = out;                                       // (T, N)
    float* log_obs_   = out + (size_t)TT * NN;                     // (T, C, N)
    float* log_hidden = out + (size_t)TT * NN + (size_t)TT * CC * NN; // (T, C, S)

    float* ws   = (float*)d_ws;
    float* wsE  = ws;                                  // C*S*N  prob emissions
    float* wsCW = wsE + (size_t)CC * SS * NN;          // T*C    prob chain weights
    float* wsH  = wsCW + (size_t)TT * CC;              // T*C*S  prob hidden states

    cw_softmax_kernel<<<TT / 256, 256, 0, stream>>>(cw_un, wsCW);
    emis_softmax_kernel<<<CC * SS, 256, 0, stream>>>(E_un, wsE);
    scan_kernel<<<1, 128, 0, stream>>>(init_un, A_un, wsH, log_hidden);

    dim3 grid((NN / 16 + 7) / 8, TT / 16);   // (79, 128)
    dim3 block(32, 8);
    emit_kernel<<<grid, block, 0, stream>>>(wsE, wsCW, wsH, log_obs, log_obs_);
}